// ESRNN_39805756899566
// MI455X (gfx1250) — compile-verified
//
#include <hip/hip_runtime.h>
#include <math.h>
#include <stdint.h>

typedef __attribute__((ext_vector_type(2))) float v2f;
typedef __attribute__((ext_vector_type(8))) float v8f;

#define SEAS   24
#define ISZ    24
#define OSZ    48
#define BB     1024
#define TT     2048
#define CC     6
#define WW     1977          // TT - OSZ - ISZ + 1
#define SFULL  2096          // 24 + 1 + 2047 + 24
#define NDD    2046          // (TT-1) - 1

// ---------------------------------------------------------------------------
// Kernel 1: per-series exponential-smoothing recurrence (latency-bound).
// One lane per series; 32 lanes/block, 32 blocks -> one wave per WGP.
// Seasonal ring buffer lives in LDS, lane-contiguous (bank-conflict free).
// ---------------------------------------------------------------------------
__global__ __launch_bounds__(32) void es_recurrence(
    const float* __restrict__ train,
    const float* __restrict__ init_lev_sms,
    const float* __restrict__ init_seas_sms,
    const float* __restrict__ init_seasonalities,
    const int*   __restrict__ idxs,
    float* __restrict__ levs,     // (B, T)
    float* __restrict__ seasf,    // (B, SFULL)
    float* __restrict__ dd_ws)    // (NDD, B) squared log-diff deltas
{
    __shared__ float sbuf[SEAS * 32];
    const int tid = threadIdx.x;
    const int b   = blockIdx.x * 32 + tid;

    const int   idx = idxs[b];
    const float la  = 1.0f / (1.0f + __expf(-init_lev_sms[idx]));
    const float sb  = 1.0f / (1.0f + __expf(-init_seas_sms[idx]));

    const float* xrow = train + (size_t)b * TT;
    float*       lrow = levs  + (size_t)b * TT;
    float*       srow = seasf + (size_t)b * SFULL;

    // seas0 = exp(init_seasonalities[idx]) -> ring buffer + seas_full[0..23]
    #pragma unroll
    for (int p = 0; p < SEAS; ++p) {
        float s = __expf(init_seasonalities[idx * SEAS + p]);
        sbuf[p * 32 + tid] = s;
        srow[p] = s;
    }
    const float s0 = sbuf[0 * 32 + tid];
    srow[SEAS] = s0;                        // seas_full[:,24] = seas0[:,0]

    float lev = xrow[0] / s0;               // lev0
    lrow[0] = lev;

    float ld_prev = 0.0f;
    for (int t = 1; t < TT; ++t) {
        if ((t & 31) == 0)                  // stream train ahead (global_prefetch_b8)
            __builtin_prefetch(xrow + t + 64, 0, 0);
        const int   pos = t % SEAS;
        const float s_i = sbuf[pos * 32 + tid];
        const float xi  = xrow[t];
        const float nl  = la * (xi / s_i) + (1.0f - la) * lev;   // new level
        const float ld  = __logf(nl / lev);
        const float ns  = sb * (xi / nl) + (1.0f - sb) * s_i;    // new seasonality
        sbuf[pos * 32 + tid] = ns;
        lrow[t]            = nl;
        srow[SEAS + t] = ns;                // seas_full[:,24+t]
        if (t >= 2) {
            const float d = ld - ld_prev;
            dd_ws[(size_t)(t - 2) * BB + b] = d * d;
        }
        ld_prev = ld;
        lev = nl;
    }

    // seas_full tail: copy columns [2048..2071] -> [2072..2095].
    // Column 24+t for t in [2024,2047] is the final write at ring slot t%24.
    #pragma unroll
    for (int j = 0; j < SEAS; ++j) {
        const int t = 2024 + j;
        srow[2072 + j] = sbuf[(t % SEAS) * 32 + tid];
    }
}

// ---------------------------------------------------------------------------
// Kernel 2: mean over batch of squared log-diff deltas, via WMMA f32 16x16x4.
// One wave computes 16 output rows: each lane pre-accumulates two 256-element
// partial sums into the A-matrix layout (lanes 0-15: K=0,1; lanes 16-31: K=2,3),
// then D = A x ones(4x16) yields the 16 row totals in column 0.
// ---------------------------------------------------------------------------
__global__ __launch_bounds__(32) void mean_sq_reduce(
    const float* __restrict__ dd,   // (NDD, B)
    float* __restrict__ out4)       // (NDD)
{
    const int lane = threadIdx.x;
    const int t0   = blockIdx.x * 16;
    const int m    = lane & 15;
    const int kh   = lane >> 4;          // 0 -> K{0,1}, 1 -> K{2,3}
    const int row  = t0 + m;

    v2f a; a.x = 0.0f; a.y = 0.0f;
    if (row < NDD) {
        const float* p0 = dd + (size_t)row * BB + (size_t)(2 * kh) * 256;
        const float* p1 = p0 + 256;
        float s0 = 0.0f, s1 = 0.0f;
        for (int j = 0; j < 256; ++j) { s0 += p0[j]; s1 += p1[j]; }
        a.x = s0; a.y = s1;
    }

    v2f ones; ones.x = 1.0f; ones.y = 1.0f;
    v8f c = {};
    // D[M][N] = sum_K A[M][K] * 1.0  (row sums, replicated across N)
    c = __builtin_amdgcn_wmma_f32_16x16x4_f32(
            false, a, false, ones, (short)0, c, false, false);

    const float inv = 1.0f / (float)BB;
    // C/D layout: lanes 0-15 hold N=lane, VGPR k = row M=k; lanes 16-31 = M=k+8.
    if (lane == 0) {
        #pragma unroll
        for (int mm = 0; mm < 8; ++mm)
            if (t0 + mm < NDD) out4[t0 + mm] = c[mm] * inv;
    } else if (lane == 16) {
        #pragma unroll
        for (int mm = 0; mm < 8; ++mm)
            if (t0 + 8 + mm < NDD) out4[t0 + 8 + mm] = c[mm] * inv;
    }
}

// ---------------------------------------------------------------------------
// Kernel 3: window builder (bandwidth-bound, ~243 MB of stores).
// Tile = 32 w x 32 b per block. All inputs staged into LDS via gfx1250
// async-to-LDS loads (ASYNCcnt path), deseasonalization done once per staged
// element, then each w emits one contiguous, fully coalesced 3.8 KB store run.
// ---------------------------------------------------------------------------
#define WTILE 32
#define BTILE 32
#define XSPAN 55             // WTILE-1 + ISZ-1 + 1 t-values per series row
#define XPAD  57             // LDS row stride (conflict avoidance)
#define LPAD  33

__device__ __forceinline__ void async_ld_b32(void* lds, const void* gaddr) {
    const uint32_t l = (uint32_t)(uintptr_t)lds;   // LDS offset = addr[31:0]
    asm volatile("global_load_async_to_lds_b32 %0, %1, off"
                 :: "v"(l), "v"(gaddr) : "memory");
}

__global__ __launch_bounds__(256) void build_windows(
    const float* __restrict__ train,     // (B, T)
    const float* __restrict__ seasf,     // (B, SFULL)
    const float* __restrict__ levs,      // (B, T)
    const float* __restrict__ info_cat,  // (B, C)
    float* __restrict__ out1)            // (W, B, ISZ+CC)
{
    __shared__ float s_x [BTILE * XPAD];   // train tile -> becomes deseas tile
    __shared__ float s_s [BTILE * XPAD];   // seas_full tile
    __shared__ float s_lv[BTILE * LPAD];   // levs tile -> becomes reciprocals
    __shared__ float s_ic[BTILE * CC];     // info_cat tile

    const int tid = threadIdx.x;
    const int w0  = blockIdx.x * WTILE;
    const int b0  = blockIdx.y * BTILE;

    // ---- stage inputs with async-to-LDS (per-lane GV addressing) ----------
    for (int e = tid; e < BTILE * XSPAN; e += 256) {
        const int r = e / XSPAN, c = e - r * XSPAN;
        async_ld_b32(&s_x[r * XPAD + c], &train[(size_t)(b0 + r) * TT + w0 + c]);
        async_ld_b32(&s_s[r * XPAD + c], &seasf[(size_t)(b0 + r) * SFULL + w0 + c]);
    }
    for (int e = tid; e < BTILE * WTILE; e += 256) {
        const int r = e >> 5, c = e & 31;
        async_ld_b32(&s_lv[r * LPAD + c],
                     &levs[(size_t)(b0 + r) * TT + (ISZ - 1) + w0 + c]);
    }
    if (tid < BTILE * CC)
        async_ld_b32(&s_ic[tid], &info_cat[b0 * CC + tid]);

    asm volatile("s_wait_asynccnt 0" ::: "memory");
    __syncthreads();

    // ---- deseasonalize once per staged element; levs -> reciprocal --------
    for (int e = tid; e < BTILE * XSPAN; e += 256) {
        const int r = e / XSPAN, c = e - r * XSPAN;
        s_x[r * XPAD + c] = s_x[r * XPAD + c] / s_s[r * XPAD + c];
    }
    for (int e = tid; e < BTILE * WTILE; e += 256) {
        const int r = e >> 5, c = e & 31;
        s_lv[r * LPAD + c] = 1.0f / s_lv[r * LPAD + c];
    }
    __syncthreads();

    // ---- emit: per w, one contiguous BTILE*30-float coalesced store run ---
    for (int wi = 0; wi < WTILE; ++wi) {
        const int w = w0 + wi;
        if (w >= WW) break;
        float* dst = out1 + ((size_t)w * BB + b0) * (ISZ + CC);
        for (int e = tid; e < BTILE * (ISZ + CC); e += 256) {
            const int bl = e / (ISZ + CC);
            const int k  = e - bl * (ISZ + CC);
            const float v = (k < ISZ)
                ? s_x[bl * XPAD + wi + k] * s_lv[bl * LPAD + wi]
                : s_ic[bl * CC + (k - ISZ)];
            dst[e] = v;
        }
    }
}

// ---------------------------------------------------------------------------
extern "C" void kernel_launch(void* const* d_in, const int* in_sizes, int n_in,
                              void* d_out, int out_size, void* d_ws, size_t ws_size,
                              hipStream_t stream) {
    const float* train              = (const float*)d_in[0];
    // d_in[1] = val, d_in[2] = test : unused by the reference outputs
    const float* info_cat           = (const float*)d_in[3];
    const float* init_lev_sms       = (const float*)d_in[4];
    const float* init_seas_sms      = (const float*)d_in[5];
    const float* init_seasonalities = (const float*)d_in[6];
    const int*   idxs               = (const int*)d_in[7];

    float* out1  = (float*)d_out;                      // input_batch (W,B,30)
    float* levs  = out1 + (size_t)WW * BB * (ISZ + CC);// (B,T)
    float* seasf = levs + (size_t)BB * TT;             // (B,SFULL)
    float* out4  = seasf + (size_t)BB * SFULL;         // (NDD)

    float* dd = (float*)d_ws;                          // (NDD, B) scratch

    es_recurrence<<<32, 32, 0, stream>>>(train, init_lev_sms, init_seas_sms,
                                         init_seasonalities, idxs,
                                         levs, seasf, dd);
    mean_sq_reduce<<<(NDD + 15) / 16, 32, 0, stream>>>(dd, out4);

    dim3 grid3((WW + WTILE - 1) / WTILE, BB / BTILE);  // 62 x 32
    build_windows<<<grid3, 256, 0, stream>>>(train, seasf, levs, info_cat, out1);
}